// SAGraphPooling_83494164234360
// MI455X (gfx1250) — compile-verified
//
#include <hip/hip_runtime.h>
#include <stdint.h>

// SAGraphPooling, mathematically reduced:
//   softmax over a size-1 axis == 1.0  =>  scoring == ones
//   stable top_k(ones, 1024)           =>  keep_indices == [0..1023] per batch
// Exact outputs:
//   Xs_out = Xs[:, :1024, :]      (B*1024*128 f32, 4 MB)
//   As_out = As[:, :1024, :1024]  (B*1024*1024 f32, 32 MB)
//   keep_values = 1.0             (B*1024 f32)
// Pure HBM-bound movement (~72 MB ~ 3.1 us @ 23.3 TB/s). The 8.6-TFLOP f32
// GEMM in the reference is dead code; computing it (slow f32 WMMA path) would
// cost 100x+ the memory-bound runtime, so it is elided.
//
// Bulk data moves through the CDNA5 async global<->LDS engine:
//   GLOBAL_LOAD_ASYNC_TO_LDS_B128 -> S_WAIT_ASYNCCNT 0 ->
//   GLOBAL_STORE_ASYNC_FROM_LDS_B128          (CDNA5 ISA s4)

// One block copies one 4 KB row: 256 lanes x 16 B (B128) per async op.
// Lane i loads into / stores from its private LDS slice, so the only
// ordering needed is the wave-level s_wait_asynccnt 0 between load-complete
// and store-issue (same-wave async loads vs stores are mutually unordered).
// S_ENDPGM's implicit wait-idle drains the pending async store.
// rows_per_batch is always a power of two -> pass log2 to avoid the ~30-op
// SALU integer-division emulation seen in round 1's prologue.
__global__ void __launch_bounds__(256) async_row_copy_kernel(
    const float* __restrict__ src, float* __restrict__ dst,
    long long src_batch_stride, long long src_row_stride, int rpb_shift)
{
    __shared__ __align__(16) float smem[1024];  // 4 KB staging tile

    const int row = blockIdx.x;
    const int b   = row >> rpb_shift;
    const int r   = row & ((1 << rpb_shift) - 1);

    const float* s = src + (size_t)b * (size_t)src_batch_stride
                         + (size_t)r * (size_t)src_row_stride
                         + (size_t)threadIdx.x * 4u;
    float* d = dst + (size_t)row * 1024u + (size_t)threadIdx.x * 4u;

    // Low 32 bits of a flat pointer into the LDS aperture == LDS byte offset
    // (CDNA5 ISA section 10.2).
    uint32_t lds_off = (uint32_t)(uintptr_t)(&smem[(size_t)threadIdx.x * 4u]);
    uint64_t gsrc    = (uint64_t)(uintptr_t)s;
    uint64_t gdst    = (uint64_t)(uintptr_t)d;

    // Memory -> LDS, async (ASYNCcnt++)
    asm volatile("global_load_async_to_lds_b128 %0, %1, off"
                 :
                 : "v"(lds_off), "v"(gsrc)
                 : "memory");
    // Load must land in LDS before the async store reads it.
    asm volatile("s_wait_asynccnt 0" ::: "memory");
    // LDS -> Memory, async (ASYNCcnt++); drained by S_ENDPGM implicit wait-idle.
    asm volatile("global_store_async_from_lds_b128 %0, %1, off"
                 :
                 : "v"(gdst), "v"(lds_off)
                 : "memory");
}

__global__ void __launch_bounds__(256) fill_ones_kernel(float* __restrict__ dst, int n)
{
    int i = blockIdx.x * blockDim.x + threadIdx.x;
    if (i < n) dst[i] = 1.0f;
}

extern "C" void kernel_launch(void* const* d_in, const int* in_sizes, int n_in,
                              void* d_out, int out_size, void* d_ws, size_t ws_size,
                              hipStream_t stream)
{
    (void)in_sizes; (void)n_in; (void)d_ws; (void)ws_size; (void)out_size;

    const float* Xs = (const float*)d_in[0];   // [B, N, F]
    const float* As = (const float*)d_in[1];   // [B, N, N]
    // d_in[2] (w0) is mathematically dead: softmax over a length-1 axis == 1.

    constexpr int B = 8, N = 2048, F = 128, K = 1024;  // keep = N/2

    float* out_xs = (float*)d_out;                      // [B, K, F]
    float* out_as = out_xs + (size_t)B * K * F;         // [B, K, K]
    float* out_kv = out_as + (size_t)B * K * K;         // [B, K]

    // Xs[:, :K, :]: the kept half of each batch is one contiguous 512 KB
    // chunk; view it as 128 rows of 1024 floats (src row stride 1024).
    async_row_copy_kernel<<<B * (K * F / 1024), 256, 0, stream>>>(
        Xs, out_xs, (long long)N * F, 1024LL, /*log2(128)=*/7);

    // As[:, :K, :K]: 8192 rows of 1024 floats, source row stride 2048.
    async_row_copy_kernel<<<B * K, 256, 0, stream>>>(
        As, out_as, (long long)N * N, (long long)N, /*log2(1024)=*/10);

    // keep_values == 1.0 everywhere.
    fill_ones_kernel<<<(B * K + 255) / 256, 256, 0, stream>>>(out_kv, B * K);
}